// BatchedLIDIA_64862596104465
// MI455X (gfx1250) — compile-verified
//
#include <hip/hip_runtime.h>
#include <hip/hip_bf16.h>
#include <stddef.h>

// ---------------------------------------------------------------------------
// LIDIA-style non-local denoiser for gfx1250 (MI455X).
// Distances computed as S(p)+S(q)-2*<patch(p),patch(q)> where the cross term
// is a Gram matrix evaluated with V_WMMA_F32_16X16X4_F32 (full f32 precision).
// Problem is tiny (~1MB live data) -> latency bound; WMMA carries the 3.2
// GFLOP distance stage. B-operand LDS loads are unconditional (zero-padding
// of K=25..27 lives only in the A operand), so the inner loop is
// 14 x ds_load_b32 + 7 x v_wmma with no EXEC predication.
// ---------------------------------------------------------------------------

typedef __attribute__((ext_vector_type(2))) float v2f;
typedef __attribute__((ext_vector_type(8))) float v8f;

#define T_ 3
#define C_ 3
#define H_ 160
#define W_ 160
#define PS_ 5
#define WSZ 29
#define KTOP 14
#define PAD_ 16          // WS2 + PS2 = 14 + 2
#define GP 192           // padded side = 160 + 2*16
#define GP2 (GP*GP)      // 36864
#define HW (H_*W_)       // 25600

// ---------------- Stage 1: per-(t,c) means of x = (noisy-0.5)/0.5 ----------
__global__ void lidia_mean_kernel(const float* __restrict__ noisy,
                                  float* __restrict__ means) {
  __shared__ float red[256];
  const int tc = blockIdx.x;                 // 0..8
  const float* p = noisy + (size_t)tc * HW;
  float s = 0.f;
  for (int i = threadIdx.x; i < HW; i += 256) s += p[i];
  red[threadIdx.x] = s;
  __syncthreads();
  for (int w = 128; w > 0; w >>= 1) {
    if (threadIdx.x < w) red[threadIdx.x] += red[threadIdx.x + w];
    __syncthreads();
  }
  if (threadIdx.x == 0) means[tc] = (red[0] / (float)HW - 0.5f) * 2.0f;
}

// ---------------- Stage 2: reflect pad -> xpad (t,3,192,192), gpad ----------
__device__ __forceinline__ int reflect160(int i) {
  i = (i < 0) ? -i : i;
  return (i >= H_) ? (2 * (H_ - 1) - i) : i;
}

__global__ void lidia_pad_kernel(const float* __restrict__ noisy,
                                 const float* __restrict__ means,
                                 float* __restrict__ xpad,
                                 float* __restrict__ gpad) {
  int idx = blockIdx.x * blockDim.x + threadIdx.x;
  if (idx >= T_ * GP2) return;
  int tt = idx / GP2;
  int r = idx % GP2;
  int y = r / GP, x = r % GP;
  int sy = reflect160(y - PAD_);
  int sx = reflect160(x - PAD_);
  const float* np = noisy + (size_t)tt * C_ * HW;
  size_t so = (size_t)sy * W_ + sx;
  float x0 = (np[so] - 0.5f) * 2.0f - means[tt * 3 + 0];
  float x1 = (np[HW + so] - 0.5f) * 2.0f - means[tt * 3 + 1];
  float x2 = (np[2 * HW + so] - 0.5f) * 2.0f - means[tt * 3 + 2];
  float* xp = xpad + (size_t)tt * C_ * GP2;
  xp[r] = x0;
  xp[GP2 + r] = x1;
  xp[2 * GP2 + r] = x2;
  gpad[(size_t)tt * GP2 + r] = 0.2989f * x0 + 0.587f * x1 + 0.114f * x2;
}

// ---------------- Stage 3: S(q) = sum over 5x5 patch of gpad^2 --------------
__global__ void lidia_sq_kernel(const float* __restrict__ gpad,
                                float* __restrict__ S) {
  int idx = blockIdx.x * blockDim.x + threadIdx.x;
  if (idx >= T_ * GP2) return;
  int tt = idx / GP2;
  int r = idx % GP2;
  int y = r / GP, x = r % GP;
  float s = 0.f;
  if (y >= 2 && y < GP - 2 && x >= 2 && x < GP - 2) {
    const float* gp = gpad + (size_t)tt * GP2;
    #pragma unroll
    for (int uy = -2; uy <= 2; ++uy)
      #pragma unroll
      for (int ux = -2; ux <= 2; ++ux) {
        float g = gp[(y + uy) * GP + (x + ux)];
        s += g * g;
      }
  }
  S[idx] = s;
}

// ---------------- Stage 4: WMMA Gram distances + top-14 + weights -----------
// One wave (32 lanes) per tile of 16 pixels (fixed row py, cols px0..px0+15).
// Lane l: m = l&15, hi = l>>4.
//   A (16x4 f32): lanes 0-15 K=0,1 ; lanes 16-31 K=2,3  (VGPR j <-> K=2*hi+j)
//   B (4x16 f32): same K striping, N = l&15
//   C (16x16 f32): VGPR j, lane l -> element (M = j + 8*hi, N = l&15)
__global__ __launch_bounds__(32)
void lidia_topk_wmma_kernel(const float* __restrict__ gpad,
                            const float* __restrict__ Sarr,
                            const float* __restrict__ wk,
                            const float* __restrict__ wb,
                            const float* __restrict__ temp,
                            float* __restrict__ aW,
                            int* __restrict__ codeW,
                            float* __restrict__ pwW) {
  __shared__ float G[33 * 52];     // gray tile: rows py-16..py+16, cols px0-18..px0+33
  __shared__ float SqL[29 * 48];   // S for candidates: dy -14..14, qx px0-16..px0+31
  __shared__ float Dtile[16 * 16]; // cross values for current 16x16 tile
  __shared__ float Md[16 * 28];    // merge scratch: per pixel, 2 sorted 14-lists
  __shared__ int Mc[16 * 28];

  const int lane = threadIdx.x;
  const int m = lane & 15;
  const int hi = lane >> 4;

  const int tile = blockIdx.x;               // 0..4799 = 3*160*10
  const int tt = tile / (H_ * 10);
  const int rem = tile % (H_ * 10);
  const int py = PAD_ + (rem / 10);          // padded row of pixels
  const int px0 = PAD_ + 16 * (rem % 10);    // padded col of first pixel

  const float* gp = gpad + (size_t)tt * GP2;
  const float* Sg = Sarr + (size_t)tt * GP2;

  // ---- stage gray tile + candidate S into LDS ----
  for (int i = lane; i < 33 * 52; i += 32) {
    int r = i / 52, c = i % 52;
    int gy = py - 16 + r;                    // always in [0,191]
    int gx = px0 - 18 + c;                   // clamp (edges only feed invalid cands)
    gx = gx < 0 ? 0 : (gx > GP - 1 ? GP - 1 : gx);
    G[i] = gp[gy * GP + gx];
  }
  for (int i = lane; i < 29 * 48; i += 32) {
    int dyi = i / 48, n = i % 48;
    int qy = py - 14 + dyi;                  // in [2,189]
    int qx = px0 - 16 + n;                   // in [0,191]
    SqL[i] = Sg[qy * GP + qx];
  }
  __syncthreads();

  // ---- per-lane loop-invariant B offsets; invalid k (25..27) clamped -------
  // B element for candidate col N=m, patch elem k, tile (dy,tcI):
  //   G[ boff[j] + dy*52 + 16*tcI ],  boff[j] = (16+uy)*52 + 2 + m + ux
  // The pixel's own patch (A operand) is the same word at boff[j] + 16 (dy=0,
  // tcI=1). Zero-padding of k>=25 lives ONLY in A, so B loads need no guard.
  int boff[14];
  #pragma unroll
  for (int j = 0; j < 14; ++j) {
    int k = (j >> 1) * 4 + 2 * hi + (j & 1);
    int kk = (k < 25) ? k : 24;              // clamped; garbage zeroed by A=0
    int uy = kk / 5 - 2, ux = kk % 5 - 2;
    boff[j] = (16 + uy) * 52 + 2 + m + ux;
  }

  // ---- build the 7 A operands once (pixel patch vectors, K padded 25->28) --
  v2f Aop[7];
  #pragma unroll
  for (int ks = 0; ks < 7; ++ks) {
    int k0 = ks * 4 + 2 * hi;
    float a0 = G[boff[2 * ks] + 16];
    float a1 = G[boff[2 * ks + 1] + 16];
    Aop[ks].x = (k0 < 25) ? a0 : 0.f;
    Aop[ks].y = (k0 + 1 < 25) ? a1 : 0.f;
  }
  const float Sp = SqL[14 * 48 + 16 + m];    // S of this lane's pixel (dy=0)

  // per-lane running top-14 (ascending), lane covers candidates n in [8*hi,8*hi+8)
  float td[KTOP];
  int tc_[KTOP];
  #pragma unroll
  for (int i = 0; i < KTOP; ++i) { td[i] = 3.0e38f; tc_[i] = 0; }

  for (int dy = -14; dy <= 14; ++dy) {
    for (int tcI = 0; tcI < 3; ++tcI) {
      const int shift = dy * 52 + 16 * tcI;  // wave-uniform
      // gather all 14 B components unconditionally, then run the WMMA chain
      float bv[14];
      #pragma unroll
      for (int j = 0; j < 14; ++j) bv[j] = G[boff[j] + shift];
      v8f acc = {};
      #pragma unroll
      for (int ks = 0; ks < 7; ++ks) {
        v2f Bop;
        Bop.x = bv[2 * ks];
        Bop.y = bv[2 * ks + 1];
        acc = __builtin_amdgcn_wmma_f32_16x16x4_f32(
            false, Aop[ks], false, Bop, (short)0, acc, false, false);
      }
      // scatter cross terms to LDS: lane holds (M = j+8*hi, N = m)
      #pragma unroll
      for (int j = 0; j < 8; ++j) Dtile[(j + 8 * hi) * 16 + m] = acc[j];
      __syncthreads();

      // scan: lane handles pixel m, candidate columns n = 8*hi + j
      #pragma unroll
      for (int j = 0; j < 8; ++j) {
        int n = 8 * hi + j;
        int dx = 16 * tcI + n - 16 - m;      // candidate qx - pixel px
        if (dx >= -14 && dx <= 14) {
          float cross = Dtile[m * 16 + n];
          float D = Sp + SqL[(dy + 14) * 48 + 16 * tcI + n] - 2.0f * cross;
          int code = (dy + 14) * WSZ + (dx + 14);
          if (D < td[KTOP - 1]) {            // branchless sorted insert
            #pragma unroll
            for (int q = KTOP - 1; q >= 0; --q) {
              float prev = (q > 0) ? td[q - 1] : -3.0e38f;
              int prevc = (q > 0) ? tc_[q - 1] : 0;
              if (D < td[q]) {
                bool fp = (D < prev);
                td[q] = fp ? prev : D;
                tc_[q] = fp ? prevc : code;
              }
            }
          }
        }
      }
      __syncthreads();   // before Dtile is overwritten next tile
    }
  }

  // ---- merge the two half-lists per pixel, then softmax / sigmoid ----------
  #pragma unroll
  for (int i = 0; i < KTOP; ++i) {
    Md[m * 28 + hi * KTOP + i] = td[i];
    Mc[m * 28 + hi * KTOP + i] = tc_[i];
  }
  __syncthreads();

  if (hi == 0) {
    float od[KTOP];
    int oc[KTOP];
    int ia = 0, ib = 0;
    #pragma unroll
    for (int i = 0; i < KTOP; ++i) {
      float da = Md[m * 28 + ia];
      float db = Md[m * 28 + KTOP + ib];
      bool ta = (da <= db);
      od[i] = ta ? da : db;
      oc[i] = ta ? Mc[m * 28 + ia] : Mc[m * 28 + KTOP + ib];
      ia += ta ? 1 : 0;
      ib += ta ? 0 : 1;
    }
    const float tmp = temp[0];
    float mx = -od[0] * tmp;
    #pragma unroll
    for (int i = 1; i < KTOP; ++i) mx = fmaxf(mx, -od[i] * tmp);
    float e[KTOP];
    float s = 0.f;
    #pragma unroll
    for (int i = 0; i < KTOP; ++i) {
      e[i] = expf(-od[i] * tmp - mx);
      s += e[i];
    }
    float invs = 1.0f / s;
    float z = wb[0];
    #pragma unroll
    for (int i = 0; i < KTOP; ++i) z += od[i] * wk[i];
    float pw = 1.0f / (1.0f + expf(-z));

    const int y = py - PAD_;
    const int x = px0 - PAD_ + m;
    const size_t pix = ((size_t)tt * H_ + y) * W_ + x;
    pwW[pix] = pw;
    #pragma unroll
    for (int i = 0; i < KTOP; ++i) {
      aW[pix * KTOP + i] = e[i] * invs;
      codeW[pix * KTOP + i] = oc[i];
    }
  }
}

// ---------------- Stage 5: gather-form aggregation + normalize --------------
__global__ void lidia_gather_kernel(const float* __restrict__ xpad,
                                    const float* __restrict__ aW,
                                    const int* __restrict__ codeW,
                                    const float* __restrict__ pwW,
                                    const float* __restrict__ means,
                                    float* __restrict__ out) {
  int idx = blockIdx.x * blockDim.x + threadIdx.x;
  if (idx >= T_ * HW) return;
  int tt = idx / HW;
  int r = idx % HW;
  int ry = r / W_, rx = r % W_;
  const float* xp = xpad + (size_t)tt * C_ * GP2;
  float n0 = 0.f, n1 = 0.f, n2 = 0.f, Z = 0.f;
  for (int vy = -2; vy <= 2; ++vy) {
    int pyy = ry + vy;
    if (pyy < 0 || pyy >= H_) continue;
    for (int vx = -2; vx <= 2; ++vx) {
      int pxx = rx + vx;
      if (pxx < 0 || pxx >= W_) continue;
      size_t pp = ((size_t)tt * H_ + pyy) * W_ + pxx;
      float pw = pwW[pp];
      Z += pw;
      const float* ap = aW + pp * KTOP;
      const int* cp = codeW + pp * KTOP;
      #pragma unroll
      for (int k = 0; k < KTOP; ++k) {
        float a = ap[k];
        int code = cp[k];
        int dy = code / WSZ - 14;
        int dx = code % WSZ - 14;
        size_t s = (size_t)(ry + PAD_ + dy) * GP + (rx + PAD_ + dx);
        float w = pw * a;
        n0 += w * xp[s];
        n1 += w * xp[GP2 + s];
        n2 += w * xp[2 * GP2 + s];
      }
    }
  }
  float inv = 1.0f / (Z + 1e-10f);
  float m0 = means[tt * 3 + 0];
  float m1 = means[tt * 3 + 1];
  float m2 = means[tt * 3 + 2];
  out[((size_t)tt * C_ + 0) * HW + r] = (n0 * inv + m0) * 0.5f + 0.5f;
  out[((size_t)tt * C_ + 1) * HW + r] = (n1 * inv + m1) * 0.5f + 0.5f;
  out[((size_t)tt * C_ + 2) * HW + r] = (n2 * inv + m2) * 0.5f + 0.5f;
}

// ---------------------------------------------------------------------------
extern "C" void kernel_launch(void* const* d_in, const int* in_sizes, int n_in,
                              void* d_out, int out_size, void* d_ws,
                              size_t ws_size, hipStream_t stream) {
  const float* noisy = (const float*)d_in[0];
  const float* wk = (const float*)d_in[1];
  const float* wb = (const float*)d_in[2];
  const float* temp = (const float*)d_in[3];
  float* out = (float*)d_out;

  // workspace layout (~11.1 MB total)
  char* ws = (char*)d_ws;
  size_t o = 0;
  float* means = (float*)(ws + o); o += 256;
  float* xpad  = (float*)(ws + o); o += (size_t)T_ * C_ * GP2 * sizeof(float);
  float* gpad  = (float*)(ws + o); o += (size_t)T_ * GP2 * sizeof(float);
  float* Sarr  = (float*)(ws + o); o += (size_t)T_ * GP2 * sizeof(float);
  float* aW    = (float*)(ws + o); o += (size_t)T_ * HW * KTOP * sizeof(float);
  int*   codeW = (int*)(ws + o);   o += (size_t)T_ * HW * KTOP * sizeof(int);
  float* pwW   = (float*)(ws + o); o += (size_t)T_ * HW * sizeof(float);
  (void)o; (void)ws_size; (void)in_sizes; (void)n_in; (void)out_size;

  lidia_mean_kernel<<<T_ * C_, 256, 0, stream>>>(noisy, means);

  int padN = T_ * GP2;
  lidia_pad_kernel<<<(padN + 255) / 256, 256, 0, stream>>>(noisy, means, xpad,
                                                           gpad);
  lidia_sq_kernel<<<(padN + 255) / 256, 256, 0, stream>>>(gpad, Sarr);

  lidia_topk_wmma_kernel<<<T_ * H_ * (W_ / 16), 32, 0, stream>>>(
      gpad, Sarr, wk, wb, temp, aW, codeW, pwW);

  int gN = T_ * HW;
  lidia_gather_kernel<<<(gN + 127) / 128, 128, 0, stream>>>(xpad, aW, codeW,
                                                            pwW, means, out);
}